// ATTNLoss_19250043420897
// MI455X (gfx1250) — compile-verified
//
#include <hip/hip_runtime.h>

#define NROWS   8192
#define ROWLEN  8192
#define RECN    (8192 * 1024)
#define REC_BLOCKS 1024
#define TOPK    32u

typedef float f4  __attribute__((ext_vector_type(4)));
typedef int   v4i __attribute__((ext_vector_type(4)));
typedef v4i __attribute__((address_space(1)))* gv4i_p;   // global
typedef v4i __attribute__((address_space(3)))* lv4i_p;   // LDS

#if defined(__has_builtin)
# if __has_builtin(__builtin_amdgcn_global_load_async_to_lds_b128)
#  define HAVE_ASYNC_B128 1
# endif
# if __has_builtin(__builtin_amdgcn_s_wait_asynccnt)
#  define HAVE_WAIT_ASYNC 1
# endif
#endif

// ---------------- Kernel A: rec_loss partial sums (streaming, NT loads) ----
__global__ __launch_bounds__(256) void rec_partial(const f4* __restrict__ x,
                                                   const f4* __restrict__ y,
                                                   float* __restrict__ part) {
  __shared__ float red[256];
  const int t = threadIdx.x;
  size_t base = (size_t)blockIdx.x * (256 * 8) + t;
  float acc = 0.f;
#pragma unroll
  for (int i = 0; i < 8; ++i) {
    size_t idx = base + (size_t)i * 256;
    f4 a = __builtin_nontemporal_load(&x[idx]);
    f4 b = __builtin_nontemporal_load(&y[idx]);
    float d0 = a.x - b.x, d1 = a.y - b.y, d2 = a.z - b.z, d3 = a.w - b.w;
    acc += d0 * d0 + d1 * d1 + d2 * d2 + d3 * d3;
  }
  red[t] = acc; __syncthreads();
  for (int s = 128; s > 0; s >>= 1) { if (t < s) red[t] += red[t + s]; __syncthreads(); }
  if (t == 0) part[blockIdx.x] = red[0];
}

// ---------------- Kernel B: per-row sum(v^2) - sum(top32 v^2) --------------
__global__ __launch_bounds__(256) void row_topk(const float* __restrict__ attn,
                                                float* __restrict__ rowpart) {
  __shared__ float    buf[ROWLEN];     // 32 KB row stage
  __shared__ unsigned hist[256];
  __shared__ unsigned sufs[257];
  __shared__ float    redf[256];
  __shared__ unsigned redu[256];
  __shared__ unsigned sel[2];

  const int t   = threadIdx.x;
  const int row = blockIdx.x;
  const float* g = attn + (size_t)row * ROWLEN;

  // ---- Stage the 32KB row into LDS: HBM is touched exactly once ----------
#ifdef HAVE_ASYNC_B128
  {
    const char* gb = (const char*)g;
    char*       lb = (char*)buf;
#pragma unroll
    for (int i = 0; i < 8; ++i) {
      int off = i * 4096 + t * 16;
      __builtin_amdgcn_global_load_async_to_lds_b128(
          (gv4i_p)(gb + off), (lv4i_p)(lb + off), 0, 0);
    }
#ifdef HAVE_WAIT_ASYNC
    __builtin_amdgcn_s_wait_asynccnt(0);
#else
    asm volatile("s_wait_asynccnt 0" ::: "memory");
#endif
  }
#else
  {
    const f4* g4 = (const f4*)g;
    f4*       b4 = (f4*)buf;
#pragma unroll
    for (int i = 0; i < 8; ++i) b4[i * 256 + t] = __builtin_nontemporal_load(&g4[i * 256 + t]);
  }
#endif
  __syncthreads();

  // ---- Build order-preserving keys in registers; accumulate sum(v^2) -----
  unsigned u[32];
  float ssq = 0.f;
#pragma unroll
  for (int i = 0; i < 32; ++i) {
    float v = buf[i * 256 + t];            // lane-strided: conflict-free
    ssq += v * v;
    unsigned b = __float_as_uint(v);
    u[i] = (b & 0x80000000u) ? ~b : (b | 0x80000000u);
  }

  // ---- 4-pass MSB radix select of the 32nd-largest key --------------------
  unsigned prefix = 0u, pmask = 0u, k = TOPK;
  for (int shift = 24; shift >= 0; shift -= 8) {
    hist[t] = 0u;
    __syncthreads();
#pragma unroll
    for (int i = 0; i < 32; ++i)
      if ((u[i] & pmask) == prefix)
        atomicAdd(&hist[(u[i] >> shift) & 255u], 1u);
    __syncthreads();

    sufs[t] = hist[t];
    if (t == 0) sufs[256] = 0u;
    __syncthreads();
    // suffix (from-the-right inclusive) scan: sufs[b] = sum_{b' >= b} hist[b']
    for (int off = 1; off < 256; off <<= 1) {
      unsigned xv = sufs[t] + ((t + off < 256) ? sufs[t + off] : 0u);
      __syncthreads();
      sufs[t] = xv;
      __syncthreads();
    }
    if (sufs[t] >= k && sufs[t + 1] < k) {   // exactly one bin satisfies this
      sel[0] = prefix | ((unsigned)t << shift);
      sel[1] = k - sufs[t + 1];
    }
    __syncthreads();
    prefix = sel[0];
    k      = sel[1];
    pmask |= (0xFFu << shift);
    __syncthreads();
  }

  // ---- prefix == exact key of the 32nd largest value ----------------------
  const unsigned U  = prefix;
  const unsigned tb = (U & 0x80000000u) ? (U ^ 0x80000000u) : ~U;
  const float    T  = __uint_as_float(tb);

  float    sgt = 0.f;
  unsigned cgt = 0u;
#pragma unroll
  for (int i = 0; i < 32; ++i) {
    if (u[i] > U) {
      unsigned vb = (u[i] & 0x80000000u) ? (u[i] ^ 0x80000000u) : ~u[i];
      float v = __uint_as_float(vb);
      sgt += v * v;
      ++cgt;
    }
  }

  redf[t] = ssq; redu[t] = cgt;
  __syncthreads();
  for (int s = 128; s > 0; s >>= 1) {
    if (t < s) { redf[t] += redf[t + s]; redu[t] += redu[t + s]; }
    __syncthreads();
  }
  const float    ssq_tot = redf[0];
  const unsigned cgt_tot = redu[0];
  __syncthreads();
  redf[t] = sgt;
  __syncthreads();
  for (int s = 128; s > 0; s >>= 1) { if (t < s) redf[t] += redf[t + s]; __syncthreads(); }

  if (t == 0) {
    float top = redf[0] + (float)(TOPK - cgt_tot) * T * T;   // tie copies at threshold
    rowpart[row] = ssq_tot - top;
  }
}

// ---------------- Kernel C: final deterministic reduction ------------------
__global__ __launch_bounds__(256) void final_reduce(const float* __restrict__ recp,
                                                    const float* __restrict__ rowp,
                                                    float* __restrict__ out) {
  __shared__ double red[256];
  const int t = threadIdx.x;
  double a = 0.0, b = 0.0;
  for (int i = t; i < REC_BLOCKS; i += 256) a += (double)recp[i];
  for (int i = t; i < NROWS;      i += 256) b += (double)rowp[i];
  red[t] = a; __syncthreads();
  for (int s = 128; s > 0; s >>= 1) { if (t < s) red[t] += red[t + s]; __syncthreads(); }
  double rec_sum = red[0];
  __syncthreads();
  red[t] = b; __syncthreads();
  for (int s = 128; s > 0; s >>= 1) { if (t < s) red[t] += red[t + s]; __syncthreads(); }
  if (t == 0) {
    double rec_loss  = rec_sum / (double)RECN;
    double attn_loss = red[0] / ((double)NROWS * (double)ROWLEN);
    out[0] = (float)(rec_loss + 0.1 * attn_loss);
  }
}

extern "C" void kernel_launch(void* const* d_in, const int* in_sizes, int n_in,
                              void* d_out, int out_size, void* d_ws, size_t ws_size,
                              hipStream_t stream) {
  (void)in_sizes; (void)n_in; (void)out_size; (void)ws_size;
  const float* x    = (const float*)d_in[0];
  const float* y    = (const float*)d_in[1];
  const float* attn = (const float*)d_in[2];
  float* ws   = (float*)d_ws;
  float* recp = ws;               // [0 .. 1023]
  float* rowp = ws + REC_BLOCKS;  // [1024 .. 9215]

  rec_partial<<<REC_BLOCKS, 256, 0, stream>>>((const f4*)x, (const f4*)y, recp);
  row_topk<<<NROWS, 256, 0, stream>>>(attn, rowp);
  final_reduce<<<1, 256, 0, stream>>>(recp, rowp, (float*)d_out);
}